// PMF_74835510165861
// MI455X (gfx1250) — compile-verified
//
#include <hip/hip_runtime.h>
#include <math.h>

#define N_USERS   10000
#define N_MOVIES  5000
#define LATENT    5
#define LAM_U     0.05f
#define LAM_V     0.05f

#define UT_TILES   (N_USERS / 16)            // 625 (exact)
#define MT_TILES   ((N_MOVIES + 15) / 16)    // 313 (last tile partial: 8 valid cols)
#define TOTAL_TILES (UT_TILES * MT_TILES)    // 195625

#define BLOCKS  4096
#define TPB     256
#define WAVES_PER_BLOCK (TPB / 32)

typedef __attribute__((ext_vector_type(2))) float v2f;
typedef __attribute__((ext_vector_type(8))) float v8f;

// ---------------------------------------------------------------------------
// Kernel 1: streaming masked-SSE with WMMA-computed low-rank scores.
// One wave32 per 16x16 tile of the (user x movie) score matrix.
//   scores = U(16x5) * V(16x5)^T   via 2x V_WMMA_F32_16X16X4_F32 (K padded 5->8)
// Branch-free OOB handling; 8 matrix loads per tile issue as one clause;
// sigmoid uses hardware v_exp_f32 + v_rcp_f32 (no IEEE divide sequence).
// ---------------------------------------------------------------------------
__global__ __launch_bounds__(TPB) void pmf_err_kernel(
    const float* __restrict__ matrix,
    const float* __restrict__ U,
    const float* __restrict__ V,
    float* __restrict__ partials)
{
    __shared__ float waveSums[WAVES_PER_BLOCK];

    const int lane   = threadIdx.x & 31;
    const int wave   = threadIdx.x >> 5;
    const int gWave  = blockIdx.x * WAVES_PER_BLOCK + wave;
    const int nWaves = gridDim.x * WAVES_PER_BLOCK;

    const int nLocal = lane & 15;   // N (or M for A operand) within tile
    const int half   = lane >> 4;   // K-half selector per ISA A/B f32 layout
    const int kBase  = half * 2;    // K offset into the 5-vector: 0 or 2

    float acc = 0.0f;

    for (int t = gWave; t < TOTAL_TILES; t += nWaves) {
        const int ut = t / MT_TILES;
        const int mt = t - ut * MT_TILES;
        const int userBase  = ut * 16;   // always in range (625*16 == 10000)
        const int movieBase = mt * 16;   // last tile partially OOB

        // ---- A operand: 16x4 f32 layout (branch-free) ----
        // lanes 0-15 : M = lane,    VGPR0 = K0, VGPR1 = K1
        // lanes 16-31: M = lane-16, VGPR0 = K2, VGPR1 = K3
        const float* up = U + (userBase + nLocal) * LATENT;
        const float ua = up[kBase];
        const float ub = up[kBase + 1];
        const float uc = up[4];
        const v2f a0 = {ua, ub};
        const v2f a1 = {half == 0 ? uc : 0.0f, 0.0f};

        // ---- B operand: 4x16 f32, mirrored layout (branch-free, clamped) ----
        const int  col   = movieBase + nLocal;
        const bool colOK = (col < N_MOVIES);
        const int  colC  = colOK ? col : (N_MOVIES - 1);   // clamped, always valid
        const float* vp = V + colC * LATENT;
        const float va = vp[kBase];
        const float vb = vp[kBase + 1];
        const float vc = vp[4];
        const v2f b0 = {colOK ? va : 0.0f, colOK ? vb : 0.0f};
        const v2f b1 = {(colOK && half == 0) ? vc : 0.0f, 0.0f};

        // ---- scores = A*B (K=8 in two K=4 steps), C/D f32 16x16 ----
        v8f c = {};
        c = __builtin_amdgcn_wmma_f32_16x16x4_f32(
                false, a0, false, b0, (short)0, c, false, false);
        c = __builtin_amdgcn_wmma_f32_16x16x4_f32(
                false, a1, false, b1, (short)0, c, false, false);

        // ---- fused sigmoid + masked SSE over this lane's 8 elements ----
        // One base address; row offsets are compile-time constants so the 8
        // loads clause together with immediate offsets (i*20000 bytes < 2^23).
        const float* mp = matrix
            + (long long)(userBase + half * 8) * N_MOVIES + colC;
        float mv[8];
        #pragma unroll
        for (int i = 0; i < 8; ++i)
            mv[i] = mp[(long long)i * N_MOVIES];

        #pragma unroll
        for (int i = 0; i < 8; ++i) {
            const float m    = mv[i];
            // sigmoid via hardware transcendentals: v_exp_f32 + v_rcp_f32
            const float pred = __builtin_amdgcn_rcpf(1.0f + __expf(-c[i]));
            const float diff = m - pred;
            // select diff (not a weight) -> single cndmask + fma, no extra mul
            const float d    = (colOK && (m != 0.0f)) ? diff : 0.0f;
            acc = fmaf(d, d, acc);
        }
    }

    // deterministic wave reduction (wave32)
    #pragma unroll
    for (int off = 16; off > 0; off >>= 1)
        acc += __shfl_down(acc, off, 32);
    if (lane == 0) waveSums[wave] = acc;
    __syncthreads();
    if (threadIdx.x == 0) {
        float s = 0.0f;
        #pragma unroll
        for (int w = 0; w < WAVES_PER_BLOCK; ++w) s += waveSums[w];
        partials[blockIdx.x] = s;
    }
}

// ---------------------------------------------------------------------------
// Kernel 2: single-block finalize. Sums block partials + regularizer norms
// in a fixed order (deterministic across graph replays), writes d_out[0].
// ---------------------------------------------------------------------------
__global__ __launch_bounds__(256) void pmf_finalize(
    const float* __restrict__ partials, int nPartials,
    const float* __restrict__ U,
    const float* __restrict__ V,
    float* __restrict__ out)
{
    __shared__ double sh[256];
    double acc = 0.0;

    for (int i = threadIdx.x; i < nPartials; i += 256)
        acc += (double)partials[i];

    for (int u = threadIdx.x; u < N_USERS; u += 256) {
        const float* p = U + u * LATENT;
        float s = p[0]*p[0] + p[1]*p[1] + p[2]*p[2] + p[3]*p[3] + p[4]*p[4];
        acc += (double)(LAM_U * sqrtf(s));
    }
    for (int m = threadIdx.x; m < N_MOVIES; m += 256) {
        const float* p = V + m * LATENT;
        float s = p[0]*p[0] + p[1]*p[1] + p[2]*p[2] + p[3]*p[3] + p[4]*p[4];
        acc += (double)(LAM_V * sqrtf(s));
    }

    sh[threadIdx.x] = acc;
    __syncthreads();
    #pragma unroll
    for (int off = 128; off > 0; off >>= 1) {
        if (threadIdx.x < off) sh[threadIdx.x] += sh[threadIdx.x + off];
        __syncthreads();
    }
    if (threadIdx.x == 0) out[0] = (float)sh[0];
}

// ---------------------------------------------------------------------------
extern "C" void kernel_launch(void* const* d_in, const int* in_sizes, int n_in,
                              void* d_out, int out_size, void* d_ws, size_t ws_size,
                              hipStream_t stream)
{
    const float* matrix = (const float*)d_in[0];   // [10000, 5000] f32
    const float* U      = (const float*)d_in[1];   // [10000, 5]    f32
    const float* V      = (const float*)d_in[2];   // [5000, 5]     f32

    float* partials = (float*)d_ws;                // BLOCKS floats (16 KB)

    pmf_err_kernel<<<BLOCKS, TPB, 0, stream>>>(matrix, U, V, partials);
    pmf_finalize<<<1, 256, 0, stream>>>(partials, BLOCKS, U, V, (float*)d_out);
}